// ElementalGTOLogNormalSkinCutoff_86723979641150
// MI455X (gfx1250) — compile-verified
//
#include <hip/hip_runtime.h>

// CDNA5 / gfx1250: wave32, V_WMMA_F32_16X16X4_F32 (f32 A/B, f32 acc).
typedef float v2f __attribute__((ext_vector_type(2)));
typedef float v8f __attribute__((ext_vector_type(8)));

namespace egto {

constexpr int NB  = 128;   // batch
constexpr int NA  = 128;   // atoms per batch
constexpr int NM  = 64;    // neighbours per atom
constexpr int FPS = 600;   // 3 (l) * 10 (bodies) * 20 (gaussians)
constexpr int WPB = 4;     // waves (atoms) per block
constexpr int WSZ = 3328;  // floats of LDS per wave region

// OFFSETS = linspace(0,6,21)[1:] = 0.3*(g+1); precomputed log(off), 1/off
constexpr float LOGOFF[20] = {
  -1.20397280f, -0.51082562f, -0.10536052f, 0.18232156f, 0.40546511f,
   0.58778666f,  0.74193734f,  0.87546874f, 0.99325177f, 1.09861229f,
   1.19392247f,  1.28093385f,  1.36097655f, 1.43508453f, 1.50407740f,
   1.56861592f,  1.62924054f,  1.68639895f, 1.74046617f, 1.79175947f };
constexpr float INVOFF[20] = {
   3.33333333f, 1.66666667f, 1.11111111f, 0.83333333f, 0.66666667f,
   0.55555556f, 0.47619048f, 0.41666667f, 0.37037037f, 0.33333333f,
   0.30303030f, 0.27777778f, 0.25641026f, 0.23809524f, 0.22222222f,
   0.20833333f, 0.19607843f, 0.18518519f, 0.17543860f, 0.16666667f };

__global__ __launch_bounds__(WPB * 32)
void egto_fp_kernel(const float* __restrict__ coords,   // [B,N,3]
                    const int*   __restrict__ charges,  // [B,N]
                    const int*   __restrict__ counts,   // [B]
                    const int*   __restrict__ neigh,    // [B,N,M]
                    float*       __restrict__ out)      // [B,N,600]
{
  __shared__ float lds[WPB * WSZ];
  const int lane = threadIdx.x & 31;
  const int wid  = threadIdx.x >> 5;
  const int atom = blockIdx.x * WPB + wid;
  const int b = atom >> 7;          // NA == 128
  const int n = atom & (NA - 1);

  float* const wbase = lds + wid * WSZ;
  float* const angT  = wbase;          // [16][64]  a-major, m contiguous
  float* const radT  = wbase + 1024;   // [32][64]  g-major, m contiguous
  float* const smsk  = wbase + 3072;   // [4][64]   species masks
  float* const tl    = wbase + 1024;   // [4][16][32] overlay (post-matmul)

  // zero pad rows: angular a=10..15, radial g=20..31
  for (int i = lane; i < 6 * 64; i += 32)  angT[10 * 64 + i] = 0.0f;
  for (int i = lane; i < 12 * 64; i += 32) radT[20 * 64 + i] = 0.0f;

  // ---- phase 1: per-neighbour elementwise (2 neighbours per lane) ----
  const float cx = coords[(b * NA + n) * 3 + 0];
  const float cy = coords[(b * NA + n) * 3 + 1];
  const float cz = coords[(b * NA + n) * 3 + 2];
  const int nbase = (b * NA + n) * NM;

  #pragma unroll
  for (int h = 0; h < 2; ++h) {
    const int m   = lane + h * 32;
    const int nb  = neigh[nbase + m];
    const int nbs = nb < 0 ? 0 : nb;
    const bool valid  = nb >= 0;
    const float coeff = valid ? 1.0f : 0.0f;
    const float lx = coords[(b * NA + nbs) * 3 + 0];
    const float ly = coords[(b * NA + nbs) * 3 + 1];
    const float lz = coords[(b * NA + nbs) * 3 + 2];
    const float x = valid ? (cx - lx) : 1.0f;   // matches reference fill
    const float y = valid ? (cy - ly) : 1.0f;
    const float z = valid ? (cz - lz) : 1.0f;

    const float d2 = x * x + y * y + z * z;
    const float d  = sqrtf(d2);
    const float dsw = (d - 1.0f) * 0.2f;        // (d-RSWITCH)/(HIGH-RSWITCH)
    const float cut = 1.0f + dsw * dsw * dsw * (-10.0f + dsw * (15.0f - 6.0f * dsw));
    const float s2  = __logf(1.0f + 2.0f / d2); // W = 2
    const float mu  = __logf(d) - 0.5f * s2;
    const float rs2 = rsqrtf(s2);
    const float pref = cut * coeff * rs2 * 0.5641895835477563f; // 1/sqrt(pi)
    const float ih   = 0.5f / s2;

    #pragma unroll
    for (int g = 0; g < 20; ++g) {
      const float e = LOGOFF[g] - mu;
      radT[g * 64 + m] = INVOFF[g] * pref * __expf(-e * e * ih);
    }

    const float id  = 1.0f / d;
    const float r2i = coeff * id * id;          // coeff / d^2   (l=0)
    const float s1  = r2i * id;                 // coeff / d^3   (l=1)
    const float s2a = r2i * id * id;            // coeff / d^4   (l=2)
    angT[0 * 64 + m] = r2i;
    angT[1 * 64 + m] = s1 * x;
    angT[2 * 64 + m] = s1 * y;
    angT[3 * 64 + m] = s1 * z;
    angT[4 * 64 + m] = s2a * x * x;
    angT[5 * 64 + m] = s2a * x * y;
    angT[6 * 64 + m] = s2a * y * y;
    angT[7 * 64 + m] = s2a * x * z;
    angT[8 * 64 + m] = s2a * y * z;
    angT[9 * 64 + m] = s2a * z * z;

    const int zq = charges[b * NA + nbs];
    smsk[0 * 64 + m] = (valid && zq == 1) ? 1.0f : 0.0f;
    smsk[1 * 64 + m] = (valid && zq == 6) ? 1.0f : 0.0f;
    smsk[2 * 64 + m] = (valid && zq == 7) ? 1.0f : 0.0f;
    smsk[3 * 64 + m] = (valid && zq == 8) ? 1.0f : 0.0f;
  }
  __syncthreads();

  // ---- phase 2: t[a,g] = sum_m angT[a,m]*mask_s[m]*radT[g,m] via WMMA ----
  // A 16x4 f32 layout: lanes 0-15/16-31 -> M=0..15; VGPR0,1 = K {0,1}/{2,3}.
  // B 4x16  f32 layout: N = lane&15;     VGPR0,1 = K {0,1}/{2,3} by lane half.
  const int row  = lane & 15;
  const int koff = (lane >> 4) << 1;   // 0 or 2

  const v8f zero8 = {0.f, 0.f, 0.f, 0.f, 0.f, 0.f, 0.f, 0.f};
  v8f acc[4][2];
  #pragma unroll
  for (int s = 0; s < 4; ++s) { acc[s][0] = zero8; acc[s][1] = zero8; }

  for (int ks = 0; ks < 16; ++ks) {
    const int kb = ks * 4 + koff;
    const v2f av = *(const v2f*)(angT + row * 64 + kb);
    const v2f b0 = *(const v2f*)(radT + row * 64 + kb);          // g tile 0
    const v2f b1 = *(const v2f*)(radT + (row + 16) * 64 + kb);   // g tile 1
    #pragma unroll
    for (int s = 0; s < 4; ++s) {
      const v2f mk = *(const v2f*)(smsk + s * 64 + kb);
      const v2f as = av * mk;
      acc[s][0] = __builtin_amdgcn_wmma_f32_16x16x4_f32(
          false, as, false, b0, (short)0, acc[s][0], false, false);
      acc[s][1] = __builtin_amdgcn_wmma_f32_16x16x4_f32(
          false, as, false, b1, (short)0, acc[s][1], false, false);
    }
  }
  __syncthreads();

  // ---- phase 3: spill D tiles to LDS (D: VGPR r -> M=r / r+8 by lane half)
  const int ah = (lane >> 4) << 3;     // 0 or 8
  #pragma unroll
  for (int s = 0; s < 4; ++s)
    #pragma unroll
    for (int t = 0; t < 2; ++t)
      #pragma unroll
      for (int r = 0; r < 8; ++r)
        tl[(s * 16 + ah + r) * 32 + row + t * 16] = acc[s][t][r];
  __syncthreads();

  // ---- phase 4: fp[l, body, g]; combos (i<j) reduce to 2*t_i*t_j ----
  const float am = (n < counts[b]) ? 1.0f : 0.0f;
  float* const op = out + (size_t)(b * NA + n) * FPS;

  for (int o = lane; o < FPS; o += 32) {
    const int l    = o / 200;
    const int rem  = o - l * 200;
    const int body = rem / 20;
    const int g    = rem - body * 20;
    const int a0 = (l == 0) ? 0 : (l == 1 ? 1 : 4);
    const int a1 = (l == 0) ? 1 : (l == 1 ? 4 : 10);
    float s = 0.0f;
    if (body < 4) {
      for (int a = a0; a < a1; ++a) {
        const float w = (a == 5 || a == 7 || a == 8) ? 2.0f : 1.0f;
        const float v = tl[(body * 16 + a) * 32 + g];
        s += w * v * v;
      }
    } else {
      const int c = body - 4;
      int i, j;
      if (c < 3)      { i = 0; j = c + 1; }
      else if (c < 5) { i = 1; j = c - 1; }
      else            { i = 2; j = 3; }
      for (int a = a0; a < a1; ++a) {
        const float w = (a == 5 || a == 7 || a == 8) ? 2.0f : 1.0f;
        s += 2.0f * w * tl[(i * 16 + a) * 32 + g] * tl[(j * 16 + a) * 32 + g];
      }
    }
    op[o] = s * am;   // LWEIGHTS are all 1.0
  }
}

} // namespace egto

extern "C" void kernel_launch(void* const* d_in, const int* in_sizes, int n_in,
                              void* d_out, int out_size, void* d_ws, size_t ws_size,
                              hipStream_t stream) {
  (void)in_sizes; (void)n_in; (void)out_size; (void)d_ws; (void)ws_size;
  const float* coords  = (const float*)d_in[0];   // [128,128,3]
  const int*   charges = (const int*)d_in[1];     // [128,128]
  const int*   counts  = (const int*)d_in[2];     // [128]
  const int*   neigh   = (const int*)d_in[3];     // [128,128,64]
  float*       out     = (float*)d_out;           // [128,128,600]

  const int blocks = (egto::NB * egto::NA) / egto::WPB;  // 4096
  egto::egto_fp_kernel<<<blocks, egto::WPB * 32, 0, stream>>>(
      coords, charges, counts, neigh, out);
}